// ST_SCGNN_64914135712512
// MI455X (gfx1250) — compile-verified
//
#include <hip/hip_runtime.h>
#include <hip/hip_bf16.h>

typedef __attribute__((ext_vector_type(2))) float v2f;
typedef __attribute__((ext_vector_type(8))) float v8f;

#define NEG_INF (-3.0e38f)

// D = A(16x4) * B(4x16) + C(16x16), fp32 WMMA (CDNA5 wave32)
static __device__ __forceinline__ v8f wmma4(v2f a, v2f b, v8f c) {
    return __builtin_amdgcn_wmma_f32_16x16x4_f32(
        /*neg_a=*/false, a, /*neg_b=*/false, b,
        /*c_mod=*/(short)0, c, /*reuse_a=*/false, /*reuse_b=*/false);
}

static __device__ __forceinline__ unsigned ballot32(bool p) {
#if __has_builtin(__builtin_amdgcn_ballot_w32)
    return __builtin_amdgcn_ballot_w32(p);
#else
    return (unsigned)__ballot(p);
#endif
}

// Low 32 bits of a generic pointer into LDS == LDS byte address (ISA §10.2).
static __device__ __forceinline__ unsigned lds_u32(const void* p) {
    return (unsigned)(unsigned long long)p;
}

// CDNA5 async global->LDS copy, 16B per lane, tracked with ASYNCcnt.
static __device__ __forceinline__ void async_copy_b128(unsigned lds_addr, const void* gaddr) {
    asm volatile("global_load_async_to_lds_b128 %0, %1, off"
                 :: "v"(lds_addr), "v"((unsigned long long)gaddr)
                 : "memory");
}
static __device__ __forceinline__ void wait_async0() {
    asm volatile("s_wait_asynccnt 0x0" ::: "memory");
}

// ---------------------------------------------------------------------------
// Stage conv + ReLU + MaxPool(1,2), writes merged-pair layout:
// xs[(n/2)][ (n&1)*32*wp + oc*wp + q ], row index n/2 in [0, 129024)
// ---------------------------------------------------------------------------
__global__ __launch_bounds__(256)
void conv_pool(const float* __restrict__ x, const float* __restrict__ cw,
               const float* __restrict__ cb, float* __restrict__ xs,
               int kw, int wp, int in_f) {
    __shared__ float sx[8 * 40];
    __shared__ float sw_[32 * 4 * 7];
    __shared__ float sb_[32];
    const int tid = threadIdx.x;
    const int nb = blockIdx.x * 8;
    for (int i = tid; i < 8 * 40; i += 256) sx[i] = x[nb * 40 + i];
    for (int i = tid; i < 32 * 4 * kw; i += 256) sw_[i] = cw[i];
    if (tid < 32) sb_[tid] = cb[tid];
    __syncthreads();

    const int slot = tid >> 5, oc = tid & 31;
    const int n = nb + slot;
    const float* xr = sx + slot * 40;
    const float* wr = sw_ + oc * 4 * kw;
    const float bias = sb_[oc];

    const int r = n / 252;           // merged graph row / 126-node group
    const int local = n - r * 252;
    const int m = local >> 1;
    const int half = local & 1;
    float* dst = xs + (size_t)(r * 126 + m) * in_f + half * 32 * wp + oc * wp;

    for (int q = 0; q < wp; ++q) {
        float v = NEG_INF;
        for (int ph = 0; ph < 2; ++ph) {
            const int p = 2 * q + ph;
            float acc = bias;
            for (int i = 0; i < 4; ++i)
                for (int j = 0; j < kw; ++j)
                    acc += xr[i * 10 + p + j] * wr[i * kw + j];
            v = fmaxf(v, acc);
        }
        dst[q] = fmaxf(v, 0.0f);     // pool(relu(a),relu(b)) == relu(max(a,b))
    }
}

// ---------------------------------------------------------------------------
// xa = tanh(xs @ bnw^T + bnb) : M=129024, N=64, K=in_f  (WMMA fp32 16x16x4)
// block = 256 thr (8 waves as 2Mx4N), covers 32 rows x 64 cols.
// A-tile staged via async global->LDS (B128); B staged transposed via DS.
// ---------------------------------------------------------------------------
__global__ __launch_bounds__(256)
void gemm_bn_tanh(const float* __restrict__ xs, const float* __restrict__ bnw,
                  const float* __restrict__ bnb, float* __restrict__ xa, int in_f) {
    extern __shared__ float smem[];
    float* sA = smem;                       // [32][in_f+4]
    float* sB = smem + 32 * (in_f + 4);     // [in_f][64]  (bnw transposed)
    const int tid = threadIdx.x;
    const int m0 = blockIdx.x * 32;
    const int lda = in_f + 4;

    __builtin_prefetch(xs + (size_t)(m0 + 32) * in_f, 0, 1);
    const int w4 = in_f >> 2;               // 16B words per A row
    for (int idx = tid; idx < 32 * w4; idx += 256) {
        int r = idx / w4, c4 = idx - r * w4;
        async_copy_b128(lds_u32(sA + r * lda + c4 * 4),
                        xs + (size_t)(m0 + r) * in_f + c4 * 4);
    }
    for (int idx = tid; idx < 64 * in_f; idx += 256) {
        int nn = idx / in_f, k = idx - nn * in_f;
        sB[k * 64 + nn] = bnw[idx];
    }
    wait_async0();
    __syncthreads();

    const int wave = tid >> 5, lane = tid & 31;
    const int wm = wave >> 2, wn = wave & 3;
    const int hi = lane >> 4, c16 = lane & 15;
    const float* aRow = sA + (16 * wm + c16) * lda + 2 * hi;
    v8f acc = {};
    for (int k0 = 0; k0 < in_f; k0 += 4) {
        v2f a; a.x = aRow[k0]; a.y = aRow[k0 + 1];
        const int kb = k0 + 2 * hi;
        v2f b; b.x = sB[kb * 64 + 16 * wn + c16];
               b.y = sB[(kb + 1) * 64 + 16 * wn + c16];
        acc = wmma4(a, b, acc);
    }
    const int n = 16 * wn + c16;
    const float bias = bnb[n];
    for (int i = 0; i < 8; ++i) {
        const int row = m0 + 16 * wm + i + 8 * hi;
        xa[(size_t)row * 64 + n] = tanhf(acc[i] + bias);
    }
}

// ---------------------------------------------------------------------------
// xg = xs @ gw : M=129024, N=32, K=in_f.  8 waves as 4Mx2N, 64 rows/block.
// Both tiles staged via async global->LDS (B128).
// ---------------------------------------------------------------------------
__global__ __launch_bounds__(256)
void gemm_xg(const float* __restrict__ xs, const float* __restrict__ gw,
             float* __restrict__ xg, int in_f) {
    extern __shared__ float smem[];
    float* sA = smem;                       // [64][in_f+4]
    float* sB = smem + 64 * (in_f + 4);     // [in_f][32] == gw layout
    const int tid = threadIdx.x;
    const int m0 = blockIdx.x * 64;
    const int lda = in_f + 4;

    __builtin_prefetch(xs + (size_t)(m0 + 64) * in_f, 0, 1);
    const int w4 = in_f >> 2;
    for (int idx = tid; idx < 64 * w4; idx += 256) {
        int r = idx / w4, c4 = idx - r * w4;
        async_copy_b128(lds_u32(sA + r * lda + c4 * 4),
                        xs + (size_t)(m0 + r) * in_f + c4 * 4);
    }
    for (int idx = tid; idx < (in_f << 3); idx += 256)   // in_f*32/4 float4s
        async_copy_b128(lds_u32(sB + idx * 4), gw + idx * 4);
    wait_async0();
    __syncthreads();

    const int wave = tid >> 5, lane = tid & 31;
    const int wm = wave >> 1, wn = wave & 1;
    const int hi = lane >> 4, c16 = lane & 15;
    const float* aRow = sA + (16 * wm + c16) * lda + 2 * hi;
    v8f acc = {};
    for (int k0 = 0; k0 < in_f; k0 += 4) {
        v2f a; a.x = aRow[k0]; a.y = aRow[k0 + 1];
        const int kb = k0 + 2 * hi;
        v2f b; b.x = sB[kb * 32 + 16 * wn + c16];
               b.y = sB[(kb + 1) * 32 + 16 * wn + c16];
        acc = wmma4(a, b, acc);
    }
    const int n = 16 * wn + c16;
    for (int i = 0; i < 8; ++i) {
        const int row = m0 + 16 * wm + i + 8 * hi;
        xg[(size_t)row * 32 + n] = acc[i];
    }
}

// ---------------------------------------------------------------------------
// Fused per-graph SCG: S=xa@xa^T (WMMA) -> softmax -> top-20 mask ->
// self-loops -> D^-1/2 A D^-1/2 -> out = A@xg + gb, ReLU (WMMA).
// One 256-thread block per graph; everything stays in LDS (~120 KB).
// xa/xg tiles arrive via async global->LDS.
// ---------------------------------------------------------------------------
__global__ __launch_bounds__(256)
void scg_adj(const float* __restrict__ xa, const float* __restrict__ xg,
             const float* __restrict__ gb, float* __restrict__ hcat, int stage) {
    extern __shared__ float smem[];
    float* sXA  = smem;                 // [128][68]
    float* sADJ = sXA + 128 * 68;       // [128][132]
    float* sXG  = sADJ + 128 * 132;     // [128][36]
    float* sDin = sXG + 128 * 36;       // [128]
    const int tid = threadIdx.x;
    const int g = blockIdx.x;
    const int wave = tid >> 5, lane = tid & 31;
    const int hi = lane >> 4, c16 = lane & 15;

    // async-stage xa (126x64) and xg (126x32) tiles, 16B units
    for (int idx = tid; idx < 126 * 16; idx += 256) {
        int r = idx >> 4, c4 = idx & 15;
        async_copy_b128(lds_u32(sXA + r * 68 + c4 * 4),
                        xa + (size_t)(g * 126 + r) * 64 + c4 * 4);
    }
    for (int idx = tid; idx < 126 * 8; idx += 256) {
        int r = idx >> 3, c4 = idx & 7;
        async_copy_b128(lds_u32(sXG + r * 36 + c4 * 4),
                        xg + (size_t)(g * 126 + r) * 32 + c4 * 4);
    }
    // zero-pad rows 126..127 (normal DS stores)
    for (int idx = tid; idx < 2 * 64; idx += 256) {
        int r = 126 + (idx >> 6);
        sXA[r * 68 + (idx & 63)] = 0.0f;
    }
    if (tid < 64) { int r = 126 + (tid >> 5); sXG[r * 36 + (tid & 31)] = 0.0f; }
    if (tid == 0) { sDin[126] = 0.0f; sDin[127] = 0.0f; }
    wait_async0();
    __syncthreads();

    // ---- Phase 1: S = xa @ xa^T (padded 128x128); wave w does row-block w
    for (int tn = 0; tn < 8; ++tn) {
        v8f acc = {};
        const float* aRow = sXA + (16 * wave + c16) * 68 + 2 * hi;
        const float* bRow = sXA + (16 * tn + c16) * 68 + 2 * hi;
        for (int k0 = 0; k0 < 64; k0 += 4) {
            v2f a; a.x = aRow[k0]; a.y = aRow[k0 + 1];
            v2f b; b.x = bRow[k0]; b.y = bRow[k0 + 1];
            acc = wmma4(a, b, acc);
        }
        for (int i = 0; i < 8; ++i)
            sADJ[(16 * wave + i + 8 * hi) * 132 + 16 * tn + c16] = acc[i];
    }
    __syncthreads();

    // ---- Phase 2: per-row softmax + top-20 + self-loop + degree
    for (int m = wave; m < 126; m += 8) {
        float v[4];
        for (int j = 0; j < 4; ++j) {
            int n = lane + 32 * j;
            v[j] = (n < 126) ? sADJ[m * 132 + n] : NEG_INF;
        }
        float mx = fmaxf(fmaxf(v[0], v[1]), fmaxf(v[2], v[3]));
        for (int off = 16; off >= 1; off >>= 1) mx = fmaxf(mx, __shfl_xor(mx, off, 32));
        float sum = 0.0f;
        for (int j = 0; j < 4; ++j) {
            int n = lane + 32 * j;
            if (n < 126) { v[j] = __expf(v[j] - mx); sum += v[j]; }
        }
        for (int off = 16; off >= 1; off >>= 1) sum += __shfl_xor(sum, off, 32);
        const float inv = 1.0f / sum;
        float w[4]; unsigned sel = 0;
        for (int j = 0; j < 4; ++j) {
            int n = lane + 32 * j;
            v[j] = (n < 126) ? v[j] * inv : NEG_INF;
            w[j] = v[j];
        }
        for (int it = 0; it < 20; ++it) {      // extract top-20 (scatter-mask)
            float bm = fmaxf(fmaxf(w[0], w[1]), fmaxf(w[2], w[3]));
            for (int off = 16; off >= 1; off >>= 1) bm = fmaxf(bm, __shfl_xor(bm, off, 32));
            int jj = -1;
            for (int j = 0; j < 4; ++j) if (jj < 0 && w[j] == bm) jj = j;
            unsigned bal = ballot32(jj >= 0);
            int src = __ffs((int)bal) - 1;
            if (lane == src) { sel |= 1u << jj; w[jj] = NEG_INF; }
        }
        float rs = 0.0f;
        for (int j = 0; j < 4; ++j) {
            int n = lane + 32 * j;
            if (n < 126) {
                float outv = ((sel >> j) & 1u) ? v[j] : 0.0f;
                if (n == m) outv = 1.0f;       // self-loop
                sADJ[m * 132 + n] = outv;
                rs += outv;
            }
        }
        for (int off = 16; off >= 1; off >>= 1) rs += __shfl_xor(rs, off, 32);
        if (lane == 0) sDin[m] = rsqrtf(fmaxf(rs, 1.0f));
    }
    __syncthreads();

    // ---- Phase 3: symmetric normalization
    for (int idx = tid; idx < 128 * 128; idx += 256) {
        int m = idx >> 7, n = idx & 127;
        sADJ[m * 132 + n] *= sDin[m] * sDin[n];
    }
    __syncthreads();

    // ---- Phase 4: out = A @ xg + gb, ReLU  (8M x 2N tiles, K=128)
    for (int tn = 0; tn < 2; ++tn) {
        v8f acc = {};
        const float* aRow = sADJ + (16 * wave + c16) * 132 + 2 * hi;
        for (int k0 = 0; k0 < 128; k0 += 4) {
            v2f a; a.x = aRow[k0]; a.y = aRow[k0 + 1];
            const int kb = k0 + 2 * hi;
            v2f b; b.x = sXG[kb * 36 + 16 * tn + c16];
                   b.y = sXG[(kb + 1) * 36 + 16 * tn + c16];
            acc = wmma4(a, b, acc);
        }
        const int n = 16 * tn + c16;
        const float bias = gb[n];
        for (int i = 0; i < 8; ++i) {
            const int m = 16 * wave + i + 8 * hi;
            if (m < 126)
                hcat[(size_t)g * (4 * 126 * 32) + stage * (126 * 32) + m * 32 + n] =
                    fmaxf(acc[i] + bias, 0.0f);
        }
    }
}

// ---------------------------------------------------------------------------
// logits = h @ lw^T + lb ; probs = softmax(logits). One wave per graph.
// ---------------------------------------------------------------------------
__global__ __launch_bounds__(32)
void lin_softmax(const float* __restrict__ h, const float* __restrict__ lw,
                 const float* __restrict__ lb, float* __restrict__ out) {
    const int r = blockIdx.x, lane = threadIdx.x;
    const float* hr = h + (size_t)r * 16128;
    float a0 = 0.0f, a1 = 0.0f, a2 = 0.0f;
    for (int k = lane; k < 16128; k += 32) {
        float v = hr[k];
        a0 += v * lw[k];
        a1 += v * lw[16128 + k];
        a2 += v * lw[2 * 16128 + k];
    }
    for (int off = 16; off >= 1; off >>= 1) {
        a0 += __shfl_xor(a0, off, 32);
        a1 += __shfl_xor(a1, off, 32);
        a2 += __shfl_xor(a2, off, 32);
    }
    if (lane == 0) {
        float l0 = a0 + lb[0], l1 = a1 + lb[1], l2 = a2 + lb[2];
        out[r * 3 + 0] = l0; out[r * 3 + 1] = l1; out[r * 3 + 2] = l2;
        float mx = fmaxf(l0, fmaxf(l1, l2));
        float e0 = __expf(l0 - mx), e1 = __expf(l1 - mx), e2 = __expf(l2 - mx);
        float inv = 1.0f / (e0 + e1 + e2);
        out[3072 + r * 3 + 0] = e0 * inv;
        out[3072 + r * 3 + 1] = e1 * inv;
        out[3072 + r * 3 + 2] = e2 * inv;
    }
}

// ---------------------------------------------------------------------------
extern "C" void kernel_launch(void* const* d_in, const int* in_sizes, int n_in,
                              void* d_out, int out_size, void* d_ws, size_t ws_size,
                              hipStream_t stream) {
    (void)in_sizes; (void)n_in; (void)out_size; (void)ws_size;
    const float* x = (const float*)d_in[0];
    const float* cw[4]  = {(const float*)d_in[3],  (const float*)d_in[5],
                           (const float*)d_in[7],  (const float*)d_in[9]};
    const float* cb[4]  = {(const float*)d_in[4],  (const float*)d_in[6],
                           (const float*)d_in[8],  (const float*)d_in[10]};
    const float* bnw[4] = {(const float*)d_in[11], (const float*)d_in[15],
                           (const float*)d_in[19], (const float*)d_in[23]};
    const float* bnb[4] = {(const float*)d_in[12], (const float*)d_in[16],
                           (const float*)d_in[20], (const float*)d_in[24]};
    const float* gw[4]  = {(const float*)d_in[13], (const float*)d_in[17],
                           (const float*)d_in[21], (const float*)d_in[25]};
    const float* gb[4]  = {(const float*)d_in[14], (const float*)d_in[18],
                           (const float*)d_in[22], (const float*)d_in[26]};
    const float* lw = (const float*)d_in[27];
    const float* lb = (const float*)d_in[28];
    float* outp = (float*)d_out;

    float* xs   = (float*)d_ws;                 // 129024*320
    float* xa   = xs + 41287680ull;             // 129024*64
    float* xg   = xa + 8257536ull;              // 129024*32
    float* hcat = xg + 4128768ull;              // 1024*16128

    const int KWs[4] = {7, 5, 3, 1};
    const int WPs[4] = {2, 3, 4, 5};
    const int FKs[4] = {128, 192, 256, 320};

    for (int s = 0; s < 4; ++s) {
        conv_pool<<<32256, 256, 0, stream>>>(x, cw[s], cb[s], xs, KWs[s], WPs[s], FKs[s]);
        size_t lds1 = (size_t)(32 * (FKs[s] + 4) + FKs[s] * 64) * sizeof(float);
        gemm_bn_tanh<<<4032, 256, lds1, stream>>>(xs, bnw[s], bnb[s], xa, FKs[s]);
        size_t lds2 = (size_t)(64 * (FKs[s] + 4) + FKs[s] * 32) * sizeof(float);
        gemm_xg<<<2016, 256, lds2, stream>>>(xs, gw[s], xg, FKs[s]);
        size_t lds3 = (size_t)(128 * 68 + 128 * 132 + 128 * 36 + 128) * sizeof(float);
        scg_adj<<<1024, 256, lds3, stream>>>(xa, xg, gb[s], hcat, s);
    }
    lin_softmax<<<1024, 32, 0, stream>>>(hcat, lw, lb, outp);
}